// GCN_87969520156747
// MI455X (gfx1250) — compile-verified
//
#include <hip/hip_runtime.h>
#include <hip/hip_bf16.h>
#include <math.h>

typedef float v2f __attribute__((ext_vector_type(2)));
typedef float v4f __attribute__((ext_vector_type(4)));
typedef float v8f __attribute__((ext_vector_type(8)));

// ---------------- elementwise / setup kernels ----------------

__global__ void zero_f32_kernel(float* __restrict__ p, long long n) {
  long long i = (long long)blockIdx.x * blockDim.x + threadIdx.x;
  long long stride = (long long)gridDim.x * blockDim.x;
  for (; i < n; i += stride) p[i] = 0.0f;
}

__global__ void init_deg_kernel(float* __restrict__ deg, int n) {
  int i = blockIdx.x * blockDim.x + threadIdx.x;
  if (i < n) deg[i] = 1.0f;  // self-loop
}

__global__ void deg_accum_kernel(const int* __restrict__ dst, float* __restrict__ deg, int E) {
  int e = blockIdx.x * blockDim.x + threadIdx.x;
  if (e < E) atomicAdd(&deg[dst[e]], 1.0f);
}

__global__ void dinv_kernel(const float* __restrict__ deg, float* __restrict__ dinv, int n) {
  int i = blockIdx.x * blockDim.x + threadIdx.x;
  if (i < n) dinv[i] = rsqrtf(deg[i]);
}

__global__ void embed_kernel(const int* __restrict__ x, const float* __restrict__ emb,
                             float* __restrict__ h, int n, int F) {
  long long idx = (long long)blockIdx.x * blockDim.x + threadIdx.x;
  long long total = (long long)n * F;
  long long stride = (long long)gridDim.x * blockDim.x;
  for (; idx < total; idx += stride) {
    int i = (int)(idx / F);
    int f = (int)(idx % F);
    h[idx] = emb[(long long)x[i] * F + f];
  }
}

// ---------------- WMMA f32 GEMM: out = A[MxK] @ W[KxN] (+bias, act) ----------------
// Each wave computes MT vertically-stacked 16x16 tiles, reusing each B fragment
// across MT WMMAs. MT is a compile-time constant -> zero divergence, no EXEC
// manipulation anywhere near v_wmma. Row-tile range starts at tmStart.
// mode: 0 = none, 1 = bias+relu, 2 = bias+sigmoid
template <int MT>
__global__ void wmma_gemm_f32_kernel(const float* __restrict__ A, const float* __restrict__ W,
                                     const float* __restrict__ bias, float* __restrict__ out,
                                     int tmStart, int mgroups, int K, int N, int mode) {
  const int wavesPerBlock = blockDim.x >> 5;
  const int wave = blockIdx.x * wavesPerBlock + ((int)threadIdx.x >> 5);
  const int ntiles = N >> 4;
  if (wave >= mgroups * ntiles) return;       // wave-uniform
  const int tg = wave / ntiles;
  const int tn = wave % ntiles;
  const int tm0 = tmStart + tg * MT;

  const int lane = (int)threadIdx.x & 31;
  const int rc   = lane & 15;            // A row (M) / B-C-D column (N)
  const int khalf = (lane >> 4) << 1;    // 0 for lanes 0-15, 2 for lanes 16-31

  const float* __restrict__ Bbase = W + (size_t)khalf * N + tn * 16 + rc;
  const float* __restrict__ Abase = A + (size_t)(tm0 * 16 + rc) * K + khalf;
  const size_t Astride = (size_t)16 * K;  // one 16-row tile down

  v8f acc[MT] = {};
  for (int k = 0; k < K; k += 4) {
    v2f b;
    b.x = Bbase[(size_t)k * N];
    b.y = Bbase[(size_t)(k + 1) * N];
    v2f a[MT];
#pragma unroll
    for (int t = 0; t < MT; ++t)
      // A 16x4 layout: VGPR0 = K=khalf, VGPR1 = K=khalf+1 (8B-aligned pair)
      a[t] = *reinterpret_cast<const v2f*>(Abase + (size_t)t * Astride + k);
#pragma unroll
    for (int t = 0; t < MT; ++t)
      acc[t] = __builtin_amdgcn_wmma_f32_16x16x4_f32(
          /*neg_a=*/false, a[t], /*neg_b=*/false, b,
          /*c_mod=*/(short)0, acc[t], /*reuse_a=*/false, /*reuse_b=*/false);
  }

  // C/D layout: VGPR r -> row r (lanes 0-15) or r+8 (lanes 16-31), col = lane&15
  const int ocol = tn * 16 + rc;
  const float bval = (mode != 0) ? bias[ocol] : 0.0f;
#pragma unroll
  for (int t = 0; t < MT; ++t) {
    const int orow0 = (tm0 + t) * 16 + ((lane >> 4) << 3);
#pragma unroll
    for (int r = 0; r < 8; ++r) {
      float v = acc[t][r] + bval;
      if (mode == 1) v = fmaxf(v, 0.0f);
      else if (mode == 2) v = 1.0f / (1.0f + __expf(-v));
      out[(size_t)(orow0 + r) * N + ocol] = v;
    }
  }
}

// scalar fallback for M % 16 remainder rows (not used when M is a multiple of 16)
__global__ void gemm_rem_kernel(const float* __restrict__ A, const float* __restrict__ W,
                                const float* __restrict__ bias, float* __restrict__ out,
                                int rowStart, int M, int K, int N, int mode) {
  int idx = blockIdx.x * blockDim.x + threadIdx.x;
  int rows = M - rowStart;
  if (idx >= rows * N) return;
  int i = rowStart + idx / N;
  int n = idx % N;
  float acc = 0.0f;
  for (int k = 0; k < K; ++k) acc += A[(size_t)i * K + k] * W[(size_t)k * N + n];
  if (mode != 0) acc += bias[n];
  if (mode == 1) acc = fmaxf(acc, 0.0f);
  else if (mode == 2) acc = 1.0f / (1.0f + __expf(-acc));
  out[(size_t)i * N + n] = acc;
}

// ---------------- edge scatter: agg[dst] += t[src] * dinv[src]*dinv[dst] ----------------
__global__ void edge_agg_kernel(const float* __restrict__ t, const int* __restrict__ src,
                                const int* __restrict__ dst, const float* __restrict__ dinv,
                                float* __restrict__ agg, int E, int F) {
  const int wavesPerBlock = blockDim.x >> 5;
  const int e = blockIdx.x * wavesPerBlock + ((int)threadIdx.x >> 5);
  if (e >= E) return;
  const int lane = (int)threadIdx.x & 31;
  const int s = src[e];
  const int d = dst[e];
  const float nrm = dinv[s] * dinv[d];
  const float* __restrict__ trow = t + (size_t)s * F;
  float* __restrict__ arow = agg + (size_t)d * F;
  if ((F & 127) == 0) {
    // 128-wide chunks: each lane gathers a float4 (global_load_b128)
    for (int f = lane * 4; f < F; f += 128) {
      v4f v = *reinterpret_cast<const v4f*>(trow + f);
      atomicAdd(&arow[f + 0], v.x * nrm);
      atomicAdd(&arow[f + 1], v.y * nrm);
      atomicAdd(&arow[f + 2], v.z * nrm);
      atomicAdd(&arow[f + 3], v.w * nrm);
    }
  } else {
    for (int f = lane; f < F; f += 32)
      atomicAdd(&arow[f], trow[f] * nrm);
  }
}

// self-loop term + bias + relu, in place into t
__global__ void combine_kernel(float* __restrict__ t, const float* __restrict__ agg,
                               const float* __restrict__ dinv, const float* __restrict__ bias,
                               int n, int F) {
  long long idx = (long long)blockIdx.x * blockDim.x + threadIdx.x;
  long long total = (long long)n * F;
  long long stride = (long long)gridDim.x * blockDim.x;
  for (; idx < total; idx += stride) {
    int i = (int)(idx / F);
    int f = (int)(idx % F);
    float di = dinv[i];
    float v = agg[idx] + t[idx] * di * di + bias[f];
    t[idx] = fmaxf(v, 0.0f);
  }
}

// ---------------- global mean pool ----------------
__global__ void pool_accum_kernel(const float* __restrict__ h, const int* __restrict__ batch,
                                  float* __restrict__ sums, float* __restrict__ cnts,
                                  int n, int F) {
  const int wavesPerBlock = blockDim.x >> 5;
  const int i = blockIdx.x * wavesPerBlock + ((int)threadIdx.x >> 5);
  if (i >= n) return;
  const int lane = (int)threadIdx.x & 31;
  const int g = batch[i];
  const float* __restrict__ hrow = h + (size_t)i * F;
  float* __restrict__ srow = sums + (size_t)g * F;
  for (int f = lane; f < F; f += 32)
    atomicAdd(&srow[f], hrow[f]);
  if (lane == 0) atomicAdd(&cnts[g], 1.0f);
}

__global__ void pool_div_kernel(const float* __restrict__ sums, const float* __restrict__ cnts,
                                float* __restrict__ g, int G, int F) {
  int idx = blockIdx.x * blockDim.x + threadIdx.x;
  if (idx >= G * F) return;
  int gi = idx / F;
  g[idx] = sums[idx] / fmaxf(cnts[gi], 1.0f);
}

// ---------------- host-side launcher ----------------
static inline int cdiv(long long a, long long b) { return (int)((a + b - 1) / b); }

static void run_gemm(const float* A, const float* W, const float* bias, float* out,
                     int M, int K, int N, int mode, hipStream_t stream) {
  const int mtiles = M >> 4;
  const int ntiles = N >> 4;
  const int mg4 = mtiles / 4;
  const int remTiles = mtiles % 4;
  if (mg4 > 0) {
    int waves = mg4 * ntiles;
    wmma_gemm_f32_kernel<4><<<cdiv(waves, 8), 256, 0, stream>>>(
        A, W, bias, out, /*tmStart=*/0, mg4, K, N, mode);
  }
  if (remTiles > 0) {
    int waves = remTiles * ntiles;
    wmma_gemm_f32_kernel<1><<<cdiv(waves, 8), 256, 0, stream>>>(
        A, W, bias, out, /*tmStart=*/mg4 * 4, remTiles, K, N, mode);
  }
  int remRows = M & 15;
  if (remRows) {
    int work = remRows * N;
    gemm_rem_kernel<<<cdiv(work, 256), 256, 0, stream>>>(A, W, bias, out, mtiles * 16, M, K, N, mode);
  }
}

extern "C" void kernel_launch(void* const* d_in, const int* in_sizes, int n_in,
                              void* d_out, int out_size, void* d_ws, size_t ws_size,
                              hipStream_t stream) {
  // inputs per setup_inputs() order
  const int*   x    = (const int*)d_in[0];      // [N]
  const int*   ei   = (const int*)d_in[1];      // [2, E]
  const int*   batch= (const int*)d_in[2];      // [N]
  const float* emb  = (const float*)d_in[3];    // [30, 80]
  const float* W1   = (const float*)d_in[4];    // [80, 128]
  const float* b1   = (const float*)d_in[5];
  const float* W2   = (const float*)d_in[6];    // [128, 128]
  const float* b2   = (const float*)d_in[7];
  const float* W3   = (const float*)d_in[8];    // [128, 256]
  const float* b3   = (const float*)d_in[9];
  const float* Wl1  = (const float*)d_in[10];   // [256, 512]
  const float* bl1  = (const float*)d_in[11];
  const float* Wl2  = (const float*)d_in[12];   // [512, 512]
  const float* bl2  = (const float*)d_in[13];
  float* out = (float*)d_out;

  const int N = in_sizes[0];
  const int E = in_sizes[1] / 2;
  const int G = out_size / 512;                 // 128 graphs
  const int EMB = 80, F1 = 128, F2 = 128, F3 = 256, FH = 512;
  const int* src = ei;
  const int* dst = ei + E;

  // workspace layout (floats), all segments 256B-aligned
  float* ws = (float*)d_ws;
  const size_t NODE_MAX = (size_t)N * 256;
  float* bufA = ws;                       // [N, up to 256]
  float* bufB = bufA + NODE_MAX;          // [N, up to 256]
  float* agg  = bufB + NODE_MAX;          // [N, up to 256]
  float* deg  = agg  + NODE_MAX;          // [N]
  float* dinv = deg  + (size_t)N;         // [N]
  float* sums = dinv + (size_t)N;         // [G, 256]
  float* cnts = sums + (size_t)G * 256;   // [G]
  float* gbuf = cnts + (size_t)G;         // [G, 256]
  float* g1   = gbuf + (size_t)G * 256;   // [G, 512]

  // degree + normalization
  init_deg_kernel<<<cdiv(N, 256), 256, 0, stream>>>(deg, N);
  deg_accum_kernel<<<cdiv(E, 256), 256, 0, stream>>>(dst, deg, E);
  dinv_kernel<<<cdiv(N, 256), 256, 0, stream>>>(deg, dinv, N);

  // embedding lookup -> bufA [N, 80]
  embed_kernel<<<cdiv((long long)N * EMB, 256), 256, 0, stream>>>(x, emb, bufA, N, EMB);

  // ---- layer 1: bufA[N,80] -> bufB[N,128] ----
  run_gemm(bufA, W1, nullptr, bufB, N, EMB, F1, 0, stream);
  zero_f32_kernel<<<4096, 256, 0, stream>>>(agg, (long long)N * F1);
  edge_agg_kernel<<<cdiv(E, 8), 256, 0, stream>>>(bufB, src, dst, dinv, agg, E, F1);
  combine_kernel<<<cdiv((long long)N * F1, 256), 256, 0, stream>>>(bufB, agg, dinv, b1, N, F1);

  // ---- layer 2: bufB[N,128] -> bufA[N,128] ----
  run_gemm(bufB, W2, nullptr, bufA, N, F1, F2, 0, stream);
  zero_f32_kernel<<<4096, 256, 0, stream>>>(agg, (long long)N * F2);
  edge_agg_kernel<<<cdiv(E, 8), 256, 0, stream>>>(bufA, src, dst, dinv, agg, E, F2);
  combine_kernel<<<cdiv((long long)N * F2, 256), 256, 0, stream>>>(bufA, agg, dinv, b2, N, F2);

  // ---- layer 3: bufA[N,128] -> bufB[N,256] ----
  run_gemm(bufA, W3, nullptr, bufB, N, F2, F3, 0, stream);
  zero_f32_kernel<<<4096, 256, 0, stream>>>(agg, (long long)N * F3);
  edge_agg_kernel<<<cdiv(E, 8), 256, 0, stream>>>(bufB, src, dst, dinv, agg, E, F3);
  combine_kernel<<<cdiv((long long)N * F3, 256), 256, 0, stream>>>(bufB, agg, dinv, b3, N, F3);

  // ---- global mean pool: bufB[N,256] -> gbuf[G,256] ----
  zero_f32_kernel<<<cdiv((long long)G * F3, 256), 256, 0, stream>>>(sums, (long long)G * F3);
  zero_f32_kernel<<<1, 256, 0, stream>>>(cnts, (long long)G);
  pool_accum_kernel<<<cdiv(N, 8), 256, 0, stream>>>(bufB, batch, sums, cnts, N, F3);
  pool_div_kernel<<<cdiv((long long)G * F3, 256), 256, 0, stream>>>(sums, cnts, gbuf, G, F3);

  // ---- MLP head ----
  run_gemm(gbuf, Wl1, bl1, g1, G, F3, FH, /*bias+relu*/1, stream);
  run_gemm(g1, Wl2, bl2, out, G, FH, FH, /*bias+sigmoid*/2, stream);
}